// BahdanauAttention_20366734917974
// MI455X (gfx1250) — compile-verified
//
#include <hip/hip_runtime.h>
#include <hip/hip_bf16.h>
#include <math.h>

#define B_     64
#define L_     2048
#define ENC_   512
#define HID_   512
#define UNITS_ 512
#define TILE_STRIDE 516   // 16x512 tile padded: 516 % 64banks = 4 -> conflict-free column reads

typedef __attribute__((ext_vector_type(2))) float v2f;
typedef __attribute__((ext_vector_type(8))) float v8f;
typedef int v4i __attribute__((vector_size(4 * sizeof(int))));

#ifndef __has_builtin
#define __has_builtin(x) 0
#endif

// ---- CDNA5 hardware tanh (V_TANH_F32) with safe fallback -----------------------------------
__device__ __forceinline__ float hw_tanh(float x) {
#if __has_builtin(__builtin_amdgcn_tanhf)
  return __builtin_amdgcn_tanhf(x);
#else
  return tanhf(x);
#endif
}

// ---- CDNA5 async global->LDS b128 (GLOBAL_LOAD_ASYNC_TO_LDS_B128, ASYNCcnt) ----------------
__device__ __forceinline__ void stage_b128(const float* __restrict__ g, float* l) {
#if __has_builtin(__builtin_amdgcn_global_load_async_to_lds_b128)
  // builtin expects v4i pointers in AS1 (global) / AS3 (LDS); cast via AS-qualified void*
  __attribute__((address_space(1))) v4i* gp =
      (__attribute__((address_space(1))) v4i*)
      (__attribute__((address_space(1))) void*)(void*)const_cast<float*>(g);
  __attribute__((address_space(3))) v4i* lp =
      (__attribute__((address_space(3))) v4i*)
      (__attribute__((address_space(3))) void*)(void*)l;
  __builtin_amdgcn_global_load_async_to_lds_b128(gp, lp, /*imm offset*/ 0, /*cpol*/ 0);
#else
  *reinterpret_cast<float4*>(l) = *reinterpret_cast<const float4*>(g);
#endif
}

__device__ __forceinline__ void wait_async_done() {
#if __has_builtin(__builtin_amdgcn_s_wait_asynccnt)
  __builtin_amdgcn_s_wait_asynccnt(0);
#else
  asm volatile("s_wait_asynccnt 0x0" ::: "memory");   // no-op if nothing async in flight
#endif
}

// ---------------- Kernel A: base[b,u] = dec_hidden[b,:]@Wh[:,u] + W_b[u] + Wc_b[u] ----------
__global__ void base_proj_kernel(const float* __restrict__ dec_hidden,
                                 const float* __restrict__ W_w,   // (HID+ENC, UNITS)
                                 const float* __restrict__ W_b,
                                 const float* __restrict__ Wc_b,
                                 float* __restrict__ base) {
  const int u = blockIdx.x * blockDim.x + threadIdx.x;   // 0..511
  const int b = blockIdx.y;
  const float* dh = dec_hidden + (size_t)b * HID_;
  float acc = 0.f;
#pragma unroll 8
  for (int h = 0; h < HID_; ++h)
    acc = fmaf(dh[h], W_w[(size_t)h * UNITS_ + u], acc);
  base[(size_t)b * UNITS_ + u] = acc + W_b[u] + Wc_b[u];
}

// ---------------- Kernel B: fused enc@We GEMM (f32 WMMA) + bias + coverage + tanh + V_w dot --
__global__ void __launch_bounds__(256, 1)
attn_score_kernel(const float* __restrict__ enc,       // (B, L, ENC)
                  const float* __restrict__ We,        // (ENC, UNITS)
                  const float* __restrict__ base,      // (B, UNITS)
                  const float* __restrict__ Wc_w,      // (UNITS)
                  const float* __restrict__ V_w,       // (UNITS)
                  const float* __restrict__ V_b,       // (1)
                  const float* __restrict__ prev_cov,  // (B, L)
                  float* __restrict__ score)           // (B, L)
{
  extern __shared__ float smem[];
  const int wave = threadIdx.x >> 5;          // 0..7, one 16-row l-tile per wave
  const int lane = threadIdx.x & 31;
  const int b  = blockIdx.x;
  const int l0 = blockIdx.y * 128 + wave * 16;

  // --- stage this wave's 16x512 enc tile into LDS (padded rows) via async b128 copies ---
  float* tile = smem + wave * (16 * TILE_STRIDE);
  const float* encRow = enc + ((size_t)b * L_ + l0) * ENC_;
  for (int i = lane; i < 16 * 128; i += 32) {
    const int r = i >> 7;
    const int c = (i & 127) << 2;
    stage_b128(encRow + (size_t)r * ENC_ + c, tile + r * TILE_STRIDE + c);
  }
  wait_async_done();
  __syncthreads();

  const int hi = lane >> 4;     // lane half selects K pair (A) / K row (B) / M+8 (C,D)
  const int lo = lane & 15;     // A row (M) and B/C/D column (N)

  float cov_r[8];
#pragma unroll
  for (int v = 0; v < 8; ++v)
    cov_r[v] = prev_cov[(size_t)b * L_ + l0 + v + 8 * hi];

  float score_acc[8];
#pragma unroll
  for (int v = 0; v < 8; ++v) score_acc[v] = 0.f;

  const float* tA = tile + lo * TILE_STRIDE + 2 * hi;

  for (int ut = 0; ut < 32; ++ut) {
    const int u0 = ut * 16;
    const float baseu = base[(size_t)b * UNITS_ + u0 + lo];
    const float wcw   = Wc_w[u0 + lo];
    const float vw    = V_w[u0 + lo];
    const float* pB   = We + (size_t)(2 * hi) * UNITS_ + u0 + lo;

    v8f acc = {0.f, 0.f, 0.f, 0.f, 0.f, 0.f, 0.f, 0.f};
#pragma unroll 8
    for (int k0 = 0; k0 < ENC_; k0 += 4) {
      // A fragment: 16x4 f32, lanes 0-15 carry K=k0,k0+1; lanes 16-31 K=k0+2,k0+3
      v2f a;  a.x  = tA[k0];                 a.y  = tA[k0 + 1];
      // B fragment: 4x16 f32, VGPR j -> K=k0+j (lo half) / k0+j+2 (hi half), N=lo
      v2f bb; bb.x = pB[(size_t)k0 * UNITS_]; bb.y = pB[(size_t)(k0 + 1) * UNITS_];
      acc = __builtin_amdgcn_wmma_f32_16x16x4_f32(false, a, false, bb,
                                                  (short)0, acc, false, false);
    }
    // epilogue on the 16x16 f32 D tile: row M = v + 8*hi, col N = u0 + lo
#pragma unroll
    for (int v = 0; v < 8; ++v) {
      const float p = acc[v] + baseu + cov_r[v] * wcw;
      score_acc[v] += hw_tanh(p) * vw;
    }
  }

  // reduce over the 16 N-lanes inside each half-wave
#pragma unroll
  for (int off = 8; off >= 1; off >>= 1)
#pragma unroll
    for (int v = 0; v < 8; ++v)
      score_acc[v] += __shfl_xor(score_acc[v], off, 32);

  if (lo == 0) {
    const float vb = V_b[0];
#pragma unroll
    for (int v = 0; v < 8; ++v)
      score[(size_t)b * L_ + l0 + v + 8 * hi] = score_acc[v] + vb;
  }
}

// ---------------- Kernel C: masked softmax over L, attention + coverage outputs -------------
__global__ void softmax_cov_kernel(const float* __restrict__ score,
                                   const int* __restrict__ mask,
                                   const float* __restrict__ prev_cov,
                                   float* __restrict__ attn,    // (B, L)
                                   float* __restrict__ cover) { // (B, L)
  __shared__ float red[256];
  const int b = blockIdx.x, tid = threadIdx.x;
  const float* s  = score + (size_t)b * L_;
  const int*   mk = mask  + (size_t)b * L_;

  float local[8];
  float m = -3.4e38f;
#pragma unroll
  for (int j = 0; j < 8; ++j) {
    const int l = j * 256 + tid;
    const float ms = s[l] + (1.0f - (float)mk[l]) * -1.0e8f;
    local[j] = ms;
    m = fmaxf(m, ms);
  }
  red[tid] = m; __syncthreads();
  for (int off = 128; off >= 1; off >>= 1) {
    if (tid < off) red[tid] = fmaxf(red[tid], red[tid + off]);
    __syncthreads();
  }
  m = red[0];
  __syncthreads();

  float sum = 0.f;
#pragma unroll
  for (int j = 0; j < 8; ++j) { local[j] = expf(local[j] - m); sum += local[j]; }
  red[tid] = sum; __syncthreads();
  for (int off = 128; off >= 1; off >>= 1) {
    if (tid < off) red[tid] += red[tid + off];
    __syncthreads();
  }
  const float inv = 1.0f / red[0];
#pragma unroll
  for (int j = 0; j < 8; ++j) {
    const int l = j * 256 + tid;
    const float a = local[j] * inv;
    attn [(size_t)b * L_ + l] = a;
    cover[(size_t)b * L_ + l] = a + prev_cov[(size_t)b * L_ + l];
  }
}

// ---------------- Kernel D: context partials over L chunks (2nd 268MB read, coalesced) ------
__global__ void context_partial_kernel(const float* __restrict__ enc,
                                       const float* __restrict__ attn,
                                       float* __restrict__ partial) { // (B, 8, ENC)
  const int b = blockIdx.x, ec = blockIdx.y, lc = blockIdx.z;
  const int e = ec * 256 + threadIdx.x;
  const float* a = attn + (size_t)b * L_;
  float acc = 0.f;
  for (int i = 0; i < 256; ++i) {
    const int l = lc * 256 + i;
    acc = fmaf(a[l], enc[((size_t)b * L_ + l) * ENC_ + e], acc);
  }
  partial[((size_t)b * 8 + lc) * ENC_ + e] = acc;
}

// ---------------- Kernel E: fold partials into context (deterministic, no atomics) ----------
__global__ void context_reduce_kernel(const float* __restrict__ partial,
                                      float* __restrict__ context) {
  const int idx = blockIdx.x * 256 + threadIdx.x;   // 0..B*ENC-1
  const int b = idx >> 9;
  const int e = idx & 511;
  float acc = 0.f;
#pragma unroll
  for (int lc = 0; lc < 8; ++lc)
    acc += partial[((size_t)b * 8 + lc) * ENC_ + e];
  context[idx] = acc;
}

extern "C" void kernel_launch(void* const* d_in, const int* in_sizes, int n_in,
                              void* d_out, int out_size, void* d_ws, size_t ws_size,
                              hipStream_t stream) {
  (void)in_sizes; (void)n_in; (void)out_size; (void)ws_size;
  const float* dec_hidden = (const float*)d_in[0];
  const float* enc_output = (const float*)d_in[1];
  const int*   enc_mask   = (const int*)  d_in[2];
  const float* prev_cov   = (const float*)d_in[3];
  const float* W_w        = (const float*)d_in[4];
  const float* W_b        = (const float*)d_in[5];
  const float* Wc_w       = (const float*)d_in[6];
  const float* Wc_b       = (const float*)d_in[7];
  const float* V_w        = (const float*)d_in[8];
  const float* V_b        = (const float*)d_in[9];

  float* out = (float*)d_out;   // [context 64*512][attn 64*2048][coverage 64*2048]
  float* ws  = (float*)d_ws;
  float* base    = ws;                              // 64*512
  float* score   = base + B_ * UNITS_;              // 64*2048
  float* partial = score + B_ * L_;                 // 64*8*512

  float* out_ctx   = out;
  float* out_attn  = out + B_ * ENC_;
  float* out_cover = out + B_ * ENC_ + B_ * L_;

  const float* We = W_w + (size_t)HID_ * UNITS_;

  base_proj_kernel<<<dim3(2, B_), 256, 0, stream>>>(dec_hidden, W_w, W_b, Wc_b, base);

  const size_t lds_bytes = (size_t)8 * 16 * TILE_STRIDE * sizeof(float);  // 264192 B < 320 KB
  attn_score_kernel<<<dim3(B_, L_ / 128), 256, lds_bytes, stream>>>(
      enc_output, We, base, Wc_w, V_w, V_b, prev_cov, score);

  softmax_cov_kernel<<<B_, 256, 0, stream>>>(score, enc_mask, prev_cov, out_attn, out_cover);

  context_partial_kernel<<<dim3(B_, ENC_ / 256, L_ / 256), 256, 0, stream>>>(
      enc_output, out_attn, partial);

  context_reduce_kernel<<<(B_ * ENC_) / 256, 256, 0, stream>>>(partial, out_ctx);
}